// ReadHead_44109314129954
// MI455X (gfx1250) — compile-verified
//
#include <hip/hip_runtime.h>
#include <hip/hip_bf16.h>

typedef float v2f __attribute__((ext_vector_type(2)));
typedef float v8f __attribute__((ext_vector_type(8)));

#define B_DIM 1024
#define N_DIM 2048
#define M_DIM 128
#define C_DIM 512
#define OC    134      // M + 6 output columns of the projection
#define OSTR  144      // row stride (floats) for o in workspace
#define EPSC  1e-8f

// ---------------------------------------------------------------------------
// Kernel A: o = x @ W^T + b   via V_WMMA_F32_16X16X4_F32
// One wave (32 threads) computes a 16x16 tile of o. grid = (B/16, ceil(OC/16)).
// ---------------------------------------------------------------------------
__global__ __launch_bounds__(32) void wmma_proj_kernel(
    const float* __restrict__ x,     // B x C
    const float* __restrict__ W,     // OC x C
    const float* __restrict__ bvec,  // OC
    float* __restrict__ o)           // B x OSTR
{
    const int row0 = blockIdx.x * 16;
    const int col0 = blockIdx.y * 16;
    const int lane = threadIdx.x & 31;
    const int half = lane >> 4;      // 0: K pair {0,1}, 1: K pair {2,3}
    const int l    = lane & 15;

    const int  col   = col0 + l;          // this lane's o-column == W row
    const bool colok = (col < OC);

    const float* xrow = x + (size_t)(row0 + l) * C_DIM;           // A row (M = l)
    const float* wrow = W + (size_t)(colok ? col : 0) * C_DIM;    // B column data

    v8f acc = {};
    for (int k0 = 0; k0 < C_DIM; k0 += 4) {
        // A 16x4: lane l holds x[row0+l][k0 + 2*half + {0,1}]
        v2f a;
        a.x = xrow[k0 + 2 * half + 0];
        a.y = xrow[k0 + 2 * half + 1];
        // B 4x16: lane holds B[k0+2*half+{0,1}][l] = W[col][k0+2*half+{0,1}]
        v2f bm;
        float b0 = wrow[k0 + 2 * half + 0];
        float b1 = wrow[k0 + 2 * half + 1];
        bm.x = colok ? b0 : 0.0f;
        bm.y = colok ? b1 : 0.0f;
        acc = __builtin_amdgcn_wmma_f32_16x16x4_f32(
            /*neg_a=*/false, a, /*neg_b=*/false, bm,
            /*c_mod=*/(short)0, acc, /*reuse_a=*/false, /*reuse_b=*/false);
    }

    if (colok) {
        const float bias = bvec[col];
        #pragma unroll
        for (int j = 0; j < 8; ++j) {
            const int row = row0 + j + 8 * half;   // D VGPR j -> M = j + 8*half
            o[(size_t)row * OSTR + col] = acc[j] + bias;
        }
    }
}

// ---------------------------------------------------------------------------
// Kernel B: fused addressing pipeline. One block (256 thr = 8 waves) per b.
//   pass1: dot/norm over mem[b]  ->  scores in LDS
//   softmax -> w_c -> w_g -> circular shift -> pow -> normalize -> w
//   pass2: r = sum_n w[n] * mem[b,n,:]
// ---------------------------------------------------------------------------
__global__ __launch_bounds__(256) void readhead_fused_kernel(
    const float* __restrict__ o,       // B x OSTR
    const float* __restrict__ w_prev,  // B x N
    const float* __restrict__ mem,     // B x N x M
    float* __restrict__ r_out,         // B x M
    float* __restrict__ w_out)         // B x N
{
    __shared__ __align__(16) float kvec[M_DIM];
    __shared__ __align__(16) float sc[N_DIM];      // scores -> exp -> w_pow -> w
    __shared__ __align__(16) float wg[N_DIM];
    __shared__ __align__(16) float red[256];
    __shared__ __align__(16) float rpart[8][M_DIM];
    __shared__ float sscal[8];                     // beta,g,s0,s1,s2,gamma,knorm

    const int b    = blockIdx.x;
    const int t    = threadIdx.x;
    const int lane = t & 31;
    const int wv   = t >> 5;

    const float* orow = o + (size_t)b * OSTR;
    if (t < M_DIM) kvec[t] = orow[t];
    if (t == 0) {
        const float ob  = orow[M_DIM + 0];
        const float og  = orow[M_DIM + 1];
        const float c0  = orow[M_DIM + 2];
        const float c1  = orow[M_DIM + 3];
        const float c2  = orow[M_DIM + 4];
        const float oga = orow[M_DIM + 5];
        const float beta  = (ob  > 20.f) ? ob  : log1pf(__expf(ob));
        const float gamma = 1.f + ((oga > 20.f) ? oga : log1pf(__expf(oga)));
        const float g     = 1.f / (1.f + __expf(-og));
        const float mx = fmaxf(c0, fmaxf(c1, c2));
        const float e0 = __expf(c0 - mx), e1 = __expf(c1 - mx), e2 = __expf(c2 - mx);
        const float es = e0 + e1 + e2;
        sscal[0] = beta; sscal[1] = g;
        sscal[2] = e0 / es; sscal[3] = e1 / es; sscal[4] = e2 / es;
        sscal[5] = gamma;
    }
    __syncthreads();

    // ||k|| via block reduce (threads 0..127 carry k^2)
    red[t] = (t < M_DIM) ? kvec[t] * kvec[t] : 0.f;
    __syncthreads();
    for (int s = 128; s > 0; s >>= 1) { if (t < s) red[t] += red[t + s]; __syncthreads(); }
    if (t == 0) sscal[6] = sqrtf(red[0]);
    __syncthreads();

    const float beta = sscal[0], g = sscal[1];
    const float s0 = sscal[2], s1 = sscal[3], s2 = sscal[4];
    const float gamma = sscal[5], knorm = sscal[6];

    // ---- pass 1: stream mem[b], per-row dot with k and sum-of-squares ----
    const float4 kq = ((const float4*)kvec)[lane];
    const float* memb = mem + (size_t)b * N_DIM * M_DIM;
    for (int n = wv; n < N_DIM; n += 8) {
        const float4 mv = ((const float4*)(memb + (size_t)n * M_DIM))[lane];
        if (n + 8 < N_DIM)
            __builtin_prefetch(memb + (size_t)(n + 8) * M_DIM + lane * 4, 0, 0);
        float d = mv.x * kq.x + mv.y * kq.y + mv.z * kq.z + mv.w * kq.w;
        float q = mv.x * mv.x + mv.y * mv.y + mv.z * mv.z + mv.w * mv.w;
        #pragma unroll
        for (int off = 16; off > 0; off >>= 1) {
            d += __shfl_xor(d, off, 32);
            q += __shfl_xor(q, off, 32);
        }
        if (lane == 0) sc[n] = beta * d / (sqrtf(q) * knorm + EPSC);
    }
    __syncthreads();

    // ---- softmax over N (in LDS) ----
    float lmax = -3.4e38f;
    for (int n = t; n < N_DIM; n += 256) lmax = fmaxf(lmax, sc[n]);
    red[t] = lmax; __syncthreads();
    for (int s = 128; s > 0; s >>= 1) { if (t < s) red[t] = fmaxf(red[t], red[t + s]); __syncthreads(); }
    const float smax = red[0];
    __syncthreads();

    float lsum = 0.f;
    for (int n = t; n < N_DIM; n += 256) { const float e = __expf(sc[n] - smax); sc[n] = e; lsum += e; }
    red[t] = lsum; __syncthreads();
    for (int s = 128; s > 0; s >>= 1) { if (t < s) red[t] += red[t + s]; __syncthreads(); }
    const float sinv = 1.f / red[0];
    __syncthreads();

    // ---- interpolate with w_prev ----
    const float* wpb = w_prev + (size_t)b * N_DIM;
    for (int n = t; n < N_DIM; n += 256)
        wg[n] = g * (sc[n] * sinv) + (1.f - g) * wpb[n];
    __syncthreads();

    // ---- circular shift + sharpen (pow) ----
    float psum = 0.f;
    for (int n = t; n < N_DIM; n += 256) {
        const int nm = (n == 0) ? (N_DIM - 1) : n - 1;
        const int np = (n == N_DIM - 1) ? 0 : n + 1;
        const float wt = s0 * wg[nm] + s1 * wg[n] + s2 * wg[np];
        const float wp = __powf(fmaxf(wt, 0.f), gamma);
        sc[n] = wp;
        psum += wp;
    }
    red[t] = psum; __syncthreads();
    for (int s = 128; s > 0; s >>= 1) { if (t < s) red[t] += red[t + s]; __syncthreads(); }
    const float pinv = 1.f / (red[0] + 1e-16f);
    __syncthreads();

    float* wob = w_out + (size_t)b * N_DIM;
    for (int n = t; n < N_DIM; n += 256) {
        const float wf = sc[n] * pinv;
        sc[n] = wf;
        wob[n] = wf;
    }
    __syncthreads();

    // ---- pass 2: r = sum_n w[n] * mem[b,n,:] ----
    float4 acc = make_float4(0.f, 0.f, 0.f, 0.f);
    for (int n = wv; n < N_DIM; n += 8) {
        const float4 mv = ((const float4*)(memb + (size_t)n * M_DIM))[lane];
        if (n + 8 < N_DIM)
            __builtin_prefetch(memb + (size_t)(n + 8) * M_DIM + lane * 4, 0, 0);
        const float wn = sc[n];
        acc.x += wn * mv.x; acc.y += wn * mv.y; acc.z += wn * mv.z; acc.w += wn * mv.w;
    }
    ((float4*)rpart[wv])[lane] = acc;
    __syncthreads();
    if (t < 32) {
        float4 s = ((float4*)rpart[0])[lane];
        #pragma unroll
        for (int i = 1; i < 8; ++i) {
            const float4 p = ((float4*)rpart[i])[lane];
            s.x += p.x; s.y += p.y; s.z += p.z; s.w += p.w;
        }
        ((float4*)(r_out + (size_t)b * M_DIM))[lane] = s;
    }
}

// ---------------------------------------------------------------------------
extern "C" void kernel_launch(void* const* d_in, const int* in_sizes, int n_in,
                              void* d_out, int out_size, void* d_ws, size_t ws_size,
                              hipStream_t stream) {
    const float* x      = (const float*)d_in[0];   // B x C
    const float* w_prev = (const float*)d_in[1];   // B x N
    const float* mem    = (const float*)d_in[2];   // B x N x M
    const float* W      = (const float*)d_in[3];   // OC x C
    const float* bvec   = (const float*)d_in[4];   // OC

    float* out   = (float*)d_out;
    float* r_out = out;                            // B*M
    float* w_out = out + (size_t)B_DIM * M_DIM;    // B*N
    float* o_ws  = (float*)d_ws;                   // B*OSTR floats (~590 KB)

    dim3 g1(B_DIM / 16, (OC + 15) / 16);
    wmma_proj_kernel<<<g1, 32, 0, stream>>>(x, W, bvec, o_ws);

    readhead_fused_kernel<<<B_DIM, 256, 0, stream>>>(o_ws, w_prev, mem, r_out, w_out);
}